// LightGCN_85727547228187
// MI455X (gfx1250) — compile-verified
//
#include <hip/hip_runtime.h>

// ---------------- problem constants (match reference) ----------------
constexpr int kNUser  = 100000;
constexpr int kNItem  = 50000;
constexpr int kNodes  = kNUser + kNItem;   // 150000
constexpr int kEmb    = 64;
constexpr int kEmbF4  = kEmb / 4;          // 16 float4 per row
constexpr int kEdges  = 4000000;           // divisible by 4
constexpr int kBatch  = 4096;
constexpr int kLayers = 3;

constexpr int kChunk       = 1024;         // edges staged to LDS per block-iteration
constexpr int kSpmmThreads = 256;          // 16 groups x 16 lanes

// ---------------- CDNA5 async global->LDS staging ----------------
#if __has_builtin(__builtin_amdgcn_global_load_async_to_lds_b128)
#define HAVE_ASYNC_LDS 1
#else
#define HAVE_ASYNC_LDS 0
#endif

typedef int v4i __attribute__((ext_vector_type(4)));
typedef __attribute__((address_space(1))) v4i gv4i;   // global int4
typedef __attribute__((address_space(3))) v4i lv4i;   // LDS int4

__device__ __forceinline__ void stage16(void* lds_dst, const void* gsrc) {
#if HAVE_ASYNC_LDS
  // (global src, lds dst, imm offset, imm cpol)
  __builtin_amdgcn_global_load_async_to_lds_b128((gv4i*)gsrc, (lv4i*)lds_dst, 0, 0);
#else
  *(int4*)lds_dst = *(const int4*)gsrc;    // synchronous fallback
#endif
}

__device__ __forceinline__ void wait_async_lds() {
#if HAVE_ASYNC_LDS
#if __has_builtin(__builtin_amdgcn_s_wait_asynccnt)
  __builtin_amdgcn_s_wait_asynccnt(0);
#else
  asm volatile("s_wait_asynccnt 0" ::: "memory");
#endif
#endif
}

__device__ __forceinline__ void atomAddF(float* p, float v) {
  // Hardware global_atomic_add_f32 (no CAS loop).
  unsafeAtomicAdd(p, v);
}

__device__ __forceinline__ float4 f4scale(float4 a, float s) {
  return make_float4(a.x * s, a.y * s, a.z * s, a.w * s);
}

// ---------------- kernel: next[row] += val * cur[col]  (segment_sum) ----------------
// First kernel in the file so the disasm snippet shows it.
// Edge metadata double-buffered through LDS via async b128 copies; each 16-lane
// group handles one edge (16 x float4 = one 64-float row).
__global__ void k_spmm(const int* __restrict__ rows, const int* __restrict__ cols,
                       const float* __restrict__ vals,
                       const float4* __restrict__ cur, float* __restrict__ nxt) {
  __shared__ __align__(16) int   s_rows[2][kChunk];
  __shared__ __align__(16) int   s_cols[2][kChunk];
  __shared__ __align__(16) float s_vals[2][kChunk];

  const int tid = threadIdx.x;
  const int q   = tid * 4;                 // each thread stages 4 ints per array
  constexpr int nChunks = (kEdges + kChunk - 1) / kChunk;

  auto stage = [&](int chunk, int bufi) {
    const int base = chunk * kChunk;
    if (base + q + 4 <= kEdges) {          // kEdges % 4 == 0 -> covers everything
      stage16(&s_rows[bufi][q], rows + base + q);
      stage16(&s_cols[bufi][q], cols + base + q);
      stage16(&s_vals[bufi][q], vals + base + q);
    }
  };

  int chunk = blockIdx.x;
  if (chunk >= nChunks) return;

  stage(chunk, 0);                         // prologue: first chunk into buffer 0
  int bufi = 0;
  const int group = tid >> 4;              // 0..15
  const int l     = tid & 15;              // float4 lane within row

  for (; chunk < nChunks; chunk += gridDim.x) {
    wait_async_lds();                      // my wave's staging for `chunk` done
    __syncthreads();                       // all waves' staging visible

    const int nextChunk = chunk + gridDim.x;
    if (nextChunk < nChunks) stage(nextChunk, bufi ^ 1);   // overlap next fetch

    const int base = chunk * kChunk;
    const int cnt  = (kEdges - base < kChunk) ? (kEdges - base) : kChunk;

#pragma unroll 2
    for (int sub = group; sub < cnt; sub += 16) {
      const int   r = s_rows[bufi][sub];
      const int   c = s_cols[bufi][sub];
      const float v = s_vals[bufi][sub];
      const float4 x = cur[c * kEmbF4 + l];        // global_load_b128, L2-resident
      float* dst = nxt + (size_t)r * kEmb + l * 4;
      atomAddF(dst + 0, v * x.x);
      atomAddF(dst + 1, v * x.y);
      atomAddF(dst + 2, v * x.z);
      atomAddF(dst + 3, v * x.w);
    }
    bufi ^= 1;
    // no trailing barrier needed: next iteration's wait+barrier protects buffers
  }
}

// ---------------- kernel: acc = cur = concat(user,item); next = 0 ----------------
__global__ void k_init(const float4* __restrict__ ue, const float4* __restrict__ ie,
                       float4* __restrict__ acc, float4* __restrict__ cur,
                       float4* __restrict__ nxt) {
  int t = blockIdx.x * blockDim.x + threadIdx.x;
  constexpr int total  = kNodes * kEmbF4;
  constexpr int userF4 = kNUser * kEmbF4;
  if (t >= total) return;
  float4 v = (t < userF4) ? ue[t] : ie[t - userF4];
  acc[t] = v;
  cur[t] = v;
  nxt[t] = make_float4(0.f, 0.f, 0.f, 0.f);
}

// ---------------- kernel: per-layer epilogue ----------------
// v = next (optionally * pop_inv); cur = v; acc += pw*v; next = 0
__global__ void k_post(float4* __restrict__ nxt, float4* __restrict__ cur,
                       float4* __restrict__ acc,
                       const float* __restrict__ upop, const float* __restrict__ ipop,
                       const float* __restrict__ pw_ptr, int apply_pop) {
  int t = blockIdx.x * blockDim.x + threadIdx.x;
  constexpr int total = kNodes * kEmbF4;
  if (t >= total) return;
  const int row = t >> 4;                  // t / kEmbF4
  float4 v = nxt[t];
  if (apply_pop) {
    const float p = (row < kNUser) ? upop[row] : ipop[row - kNUser];
    v = f4scale(v, p);
  }
  const float pw = pw_ptr[0];
  cur[t] = v;
  float4 a = acc[t];
  a.x += pw * v.x; a.y += pw * v.y; a.z += pw * v.z; a.w += pw * v.w;
  acc[t] = a;
  nxt[t] = make_float4(0.f, 0.f, 0.f, 0.f);  // ready for next layer
}

// ---------------- kernel: six batched gathers ----------------
__global__ void k_gather(const float4* __restrict__ acc,
                         const float4* __restrict__ ue, const float4* __restrict__ ie,
                         const int* __restrict__ users, const int* __restrict__ pos,
                         const int* __restrict__ neg, float4* __restrict__ out) {
  int t = blockIdx.x * blockDim.x + threadIdx.x;
  constexpr int total = 6 * kBatch * kEmbF4;
  if (t >= total) return;
  const int l  = t & (kEmbF4 - 1);
  const int r  = t >> 4;
  const int b  = r >> 12;                  // r / kBatch
  const int rr = r & (kBatch - 1);
  float4 o;
  if (b == 0)      { int i = users[rr]; o = f4scale(acc[i * kEmbF4 + l],            0.25f); }
  else if (b == 1) { int i = pos[rr];   o = f4scale(acc[(kNUser + i) * kEmbF4 + l], 0.25f); }
  else if (b == 2) { int i = neg[rr];   o = f4scale(acc[(kNUser + i) * kEmbF4 + l], 0.25f); }
  else if (b == 3) { int i = users[rr]; o = ue[i * kEmbF4 + l]; }
  else if (b == 4) { int i = pos[rr];   o = ie[i * kEmbF4 + l]; }
  else             { int i = neg[rr];   o = ie[i * kEmbF4 + l]; }
  out[t] = o;
}

// ---------------- host launcher ----------------
extern "C" void kernel_launch(void* const* d_in, const int* in_sizes, int n_in,
                              void* d_out, int out_size, void* d_ws, size_t ws_size,
                              hipStream_t stream) {
  (void)in_sizes; (void)n_in; (void)out_size; (void)ws_size;

  const float* ue   = (const float*)d_in[0];
  const float* ie   = (const float*)d_in[1];
  const int*   er   = (const int*)d_in[2];
  const int*   ec   = (const int*)d_in[3];
  const float* ev   = (const float*)d_in[4];
  const float* upop = (const float*)d_in[5];
  const float* ipop = (const float*)d_in[6];
  const float* pw   = (const float*)d_in[7];
  const int*   users = (const int*)d_in[8];
  const int*   pos   = (const int*)d_in[9];
  const int*   neg   = (const int*)d_in[10];

  float* ws = (float*)d_ws;
  const size_t nodeF = (size_t)kNodes * kEmb;
  float* acc = ws;
  float* cur = ws + nodeF;
  float* nxt = ws + 2 * nodeF;             // 3 * 38.4 MB = 115.2 MB scratch

  const int totalF4 = kNodes * kEmbF4;
  k_init<<<(totalF4 + 255) / 256, 256, 0, stream>>>(
      (const float4*)ue, (const float4*)ie, (float4*)acc, (float4*)cur, (float4*)nxt);

  const int nChunks = (kEdges + kChunk - 1) / kChunk;
  const int spmmBlocks = (nChunks < 1024) ? nChunks : 1024;
  for (int k = 0; k < kLayers; ++k) {
    k_spmm<<<spmmBlocks, kSpmmThreads, 0, stream>>>(er, ec, ev, (const float4*)cur, nxt);
    k_post<<<(totalF4 + 255) / 256, 256, 0, stream>>>(
        (float4*)nxt, (float4*)cur, (float4*)acc, upop, ipop, pw, (k == 0) ? 1 : 0);
  }

  k_gather<<<(6 * kBatch * kEmbF4 + 255) / 256, 256, 0, stream>>>(
      (const float4*)acc, (const float4*)ue, (const float4*)ie,
      users, pos, neg, (float4*)d_out);
}